// Prototypes_3204045603073
// MI455X (gfx1250) — compile-verified
//
#include <hip/hip_runtime.h>
#include <hip/hip_bf16.h>

// ---------------------------------------------------------------------------
// Prototypes / nearest-patch kernel for MI455X (gfx1250, wave32, WMMA).
//
// B=8, N=16384, P=512, D=64.
//   pn   = normalize(prototypes)                 (P,D)
//   xp   = x . pn^T                              (GEMM M=B*N, N=P, K=64)
//   dist = sqrt(max(|x|^2 - 2 xp + |pn|^2,0)+1e-8)
//   out  = (min over N, argmin over N)           (B,P) each
//
// Roofline: 8.6 GFLOP fp32 GEMM vs 32MB traffic (1.4us @ 23.3TB/s) ->
// matrix-core bound; use V_WMMA_F32_16X16X4_F32 chains (exact fp32 path).
// Reduction fused via packed (f32bits<<32 | row_idx) u64 atomic-min keys:
// monotone for dist>=0, ties resolve to smallest index (argmin semantics),
// and min-combining is order independent => deterministic across replays.
// ---------------------------------------------------------------------------

typedef __attribute__((ext_vector_type(2))) float v2f;
typedef __attribute__((ext_vector_type(4))) float v4f;
typedef __attribute__((ext_vector_type(8))) float v8f;

#define B_      8
#define N_      16384
#define P_      512
#define D_      64
#define ROWS    128          // x rows staged per workgroup
#define THREADS 512          // 16 waves of 32
#define CT      2            // col-tiles (of 16 prototypes) per wave: 16*32=512

// ---------------- kernel 1: reset the atomic key buffer --------------------
__global__ void init_keys_kernel(unsigned long long* __restrict__ keys) {
  int i = blockIdx.x * blockDim.x + threadIdx.x;
  if (i < B_ * P_) keys[i] = 0xFFFFFFFFFFFFFFFFull;
}

// ---------------- kernel 2: normalize prototypes ---------------------------
__global__ void proto_prep_kernel(const float* __restrict__ proto,
                                  float* __restrict__ pn,
                                  float* __restrict__ p2) {
  int p = blockIdx.x * blockDim.x + threadIdx.x;
  if (p >= P_) return;
  const float* src = proto + (size_t)p * D_;
  float v[D_];
  float s = 0.f;
#pragma unroll
  for (int i = 0; i < D_; i += 4) {
    v4f t = *(const v4f*)(src + i);
    v[i + 0] = t.x; v[i + 1] = t.y; v[i + 2] = t.z; v[i + 3] = t.w;
    s += t.x * t.x + t.y * t.y + t.z * t.z + t.w * t.w;
  }
  float nrm = sqrtf(s);
  float inv = 1.0f / fmaxf(nrm, 1e-12f);
  float* dst = pn + (size_t)p * D_;
  float s2 = 0.f;
#pragma unroll
  for (int i = 0; i < D_; i += 4) {
    v4f t;
    t.x = v[i + 0] * inv; t.y = v[i + 1] * inv;
    t.z = v[i + 2] * inv; t.w = v[i + 3] * inv;
    s2 += t.x * t.x + t.y * t.y + t.z * t.z + t.w * t.w;
    *(v4f*)(dst + i) = t;
  }
  p2[p] = s2;
}

// ---------------- kernel 3: fused WMMA GEMM + min/argmin -------------------
__launch_bounds__(THREADS, 1)
__global__ void proto_dist_min_kernel(const float* __restrict__ x,
                                      const float* __restrict__ pn,
                                      const float* __restrict__ p2,
                                      unsigned long long* __restrict__ keys) {
  __shared__ float sX[ROWS * D_];   // 32 KB
  __shared__ float sX2[ROWS];

  const int b     = blockIdx.y;
  const int chunk = blockIdx.x * ROWS;                 // base row in N
  const float* xsrc = x + ((size_t)b * N_ + chunk) * D_;

  // Cooperative coalesced staging of the 128x64 x tile (float4).
  for (int i = threadIdx.x; i < ROWS * D_ / 4; i += THREADS)
    ((v4f*)sX)[i] = ((const v4f*)xsrc)[i];
  __syncthreads();

  // Per-row |x|^2.
  if (threadIdx.x < ROWS) {
    const float* r = sX + threadIdx.x * D_;
    float s = 0.f;
#pragma unroll
    for (int i = 0; i < D_; ++i) s += r[i] * r[i];
    sX2[threadIdx.x] = s;
  }
  __syncthreads();

  const int wave = threadIdx.x >> 5;      // 0..15
  const int lane = threadIdx.x & 31;
  const int lh   = lane & 15;             // position within half-wave
  const int hi   = lane >> 4;             // 0: K/M low half, 1: high half

  // Preload this wave's B fragments (32 prototype columns) into VGPRs.
  // A-matrix 16x4 f32 layout (ISA 7.12.2): lanes 0-15 hold {K=4k,4k+1},
  // lanes 16-31 hold {K=4k+2,4k+3}; B (4x16) mirrors with lane = column.
  const int pbase = wave * (CT * 16);
  v2f   bfrag[CT][16];
  float p2reg[CT];
#pragma unroll
  for (int ct = 0; ct < CT; ++ct) {
    const int col = pbase + ct * 16 + lh;
    const float* prow = pn + (size_t)col * D_;
#pragma unroll
    for (int k = 0; k < 16; ++k)
      bfrag[ct][k] = *(const v2f*)(prow + 4 * k + 2 * hi);
    p2reg[ct] = p2[col];
  }

  unsigned long long runKey[CT];
#pragma unroll
  for (int ct = 0; ct < CT; ++ct) runKey[ct] = 0xFFFFFFFFFFFFFFFFull;

#pragma unroll 1
  for (int rt = 0; rt < ROWS / 16; ++rt) {
    const int mbase = rt * 16;

    // A fragments for this 16-row tile, from LDS.
    v2f afrag[16];
#pragma unroll
    for (int k = 0; k < 16; ++k)
      afrag[k] = *(const v2f*)(sX + (mbase + lh) * D_ + 4 * k + 2 * hi);

    v8f acc[CT] = {};
    // Two independent 16-deep WMMA chains (K=64 total).
#pragma unroll
    for (int k = 0; k < 16; ++k) {
#pragma unroll
      for (int ct = 0; ct < CT; ++ct) {
        acc[ct] = __builtin_amdgcn_wmma_f32_16x16x4_f32(
            /*neg_a=*/false, afrag[k], /*neg_b=*/false, bfrag[ct][k],
            /*c_mod=*/(short)0, acc[ct], /*reuse_a=*/false, /*reuse_b=*/false);
      }
    }

    // Epilogue: distance + running (dist,idx) min.
    // C/D layout: VGPR r -> M=r (lanes 0-15) / M=8+r (lanes 16-31), N=lane%16.
#pragma unroll
    for (int ct = 0; ct < CT; ++ct) {
#pragma unroll
      for (int r = 0; r < 8; ++r) {
        const int rowLocal = mbase + hi * 8 + r;
        float sq = sX2[rowLocal] - 2.0f * acc[ct][r] + p2reg[ct];
        sq = fmaxf(sq, 0.0f);
        const float d = sqrtf(sq + 1e-8f);
        const unsigned long long k64 =
            ((unsigned long long)__float_as_uint(d) << 32) |
            (unsigned int)(chunk + rowLocal);
        runKey[ct] = (k64 < runKey[ct]) ? k64 : runKey[ct];
      }
    }
  }

  // Combine the two lane halves (same column, different M rows), then one
  // u64 atomic-min per prototype column.
#pragma unroll
  for (int ct = 0; ct < CT; ++ct) {
    unsigned long long o =
        (unsigned long long)__shfl_xor((unsigned long long)runKey[ct], 16, 32);
    runKey[ct] = (o < runKey[ct]) ? o : runKey[ct];
    if (hi == 0) {
      const int col = pbase + ct * 16 + lh;
      atomicMin(&keys[(size_t)b * P_ + col], runKey[ct]);
    }
  }
}

// ---------------- kernel 4: decode keys into outputs -----------------------
__global__ void finalize_kernel(const unsigned long long* __restrict__ keys,
                                float* __restrict__ out) {
  int i = blockIdx.x * blockDim.x + threadIdx.x;
  if (i >= B_ * P_) return;
  const unsigned long long k = keys[i];
  out[i] = __uint_as_float((unsigned int)(k >> 32));            // min dist
  out[B_ * P_ + i] = (float)(unsigned int)(k & 0xFFFFFFFFu);    // argmin idx
}

// ---------------------------------------------------------------------------
extern "C" void kernel_launch(void* const* d_in, const int* in_sizes, int n_in,
                              void* d_out, int out_size, void* d_ws, size_t ws_size,
                              hipStream_t stream) {
  (void)in_sizes; (void)n_in; (void)out_size; (void)ws_size;
  const float* x     = (const float*)d_in[0];   // (B,N,D) f32
  const float* proto = (const float*)d_in[1];   // (P,D)   f32
  float* out = (float*)d_out;                   // [B*P dist | B*P idx]

  // Workspace layout (8-byte aligned throughout):
  //   [0,            P*D*4)          normalized prototypes
  //   [P*D*4,        P*D*4 + P*4)    p2
  //   [P*D*4 + P*4,  ... + B*P*8)    u64 (dist,idx) keys
  char* ws = (char*)d_ws;
  float* pn = (float*)ws;
  float* p2 = (float*)(ws + (size_t)P_ * D_ * sizeof(float));
  unsigned long long* keys =
      (unsigned long long*)(ws + (size_t)P_ * D_ * sizeof(float)
                               + (size_t)P_ * sizeof(float));

  init_keys_kernel<<<(B_ * P_ + 255) / 256, 256, 0, stream>>>(keys);
  proto_prep_kernel<<<(P_ + 255) / 256, 256, 0, stream>>>(proto, pn, p2);
  dim3 grid(N_ / ROWS, B_);
  proto_dist_min_kernel<<<grid, THREADS, 0, stream>>>(x, pn, p2, keys);
  finalize_kernel<<<(B_ * P_ + 255) / 256, 256, 0, stream>>>(keys, out);
}